// GCNModel_22230750724231
// MI455X (gfx1250) — compile-verified
//
#include <hip/hip_runtime.h>
#include <hip/hip_bf16.h>

typedef __attribute__((ext_vector_type(2))) float v2f;
typedef __attribute__((ext_vector_type(8))) float v8f;

// ---------------------------------------------------------------------------
// Generic zero-fill
// ---------------------------------------------------------------------------
__global__ __launch_bounds__(256) void zero_kernel(float* __restrict__ p, int n) {
    int i = blockIdx.x * blockDim.x + threadIdx.x;
    if (i < n) p[i] = 0.0f;
}

// ---------------------------------------------------------------------------
// Degree (with self loop) -> dinv = rsqrt(deg + 1)
// ---------------------------------------------------------------------------
__global__ __launch_bounds__(256) void deg_kernel(const int* __restrict__ dst,
                                                  float* __restrict__ deg, int E) {
    int i = blockIdx.x * blockDim.x + threadIdx.x;
    if (i < E) atomicAdd(&deg[dst[i]], 1.0f);
}

__global__ __launch_bounds__(256) void dinv_kernel(float* __restrict__ deg, int n) {
    int i = blockIdx.x * blockDim.x + threadIdx.x;
    if (i < n) deg[i] = rsqrtf(deg[i] + 1.0f);
}

// ---------------------------------------------------------------------------
// WMMA fp32 GEMM:  C[M x N] = A[M x K] @ W[K x N] (+bias) (+relu)
// Requires M % 32 == 0, N == 256, K % 4 == 0.
// Block = 256 threads (8 waves); block tile = 32 rows x 256 cols.
// Wave w owns a 32x32 patch (2x2 grid of 16x16 WMMA tiles) at cols w*32.
// A fragments reused across 2 col tiles; B fragments across 2 row tiles:
// 4 WMMAs per (2 LDS b64 + 4 global b32) loads.
// A tile (32 x K) staged in LDS, stride K+4 -> 16-lane reads hit distinct banks.
// ---------------------------------------------------------------------------
__global__ __launch_bounds__(256) void gemm_wmma_f32(
    const float* __restrict__ A, const float* __restrict__ W,
    const float* __restrict__ bias, float* __restrict__ C,
    int M, int N, int K, int relu) {
    __shared__ float As[32 * 260];           // up to K=256, row stride K+4
    const int stride = K + 4;
    const int m0   = blockIdx.x * 32;
    const int wave = threadIdx.x >> 5;
    const int lane = threadIdx.x & 31;
    const int n0   = wave * 32;              // 8 waves * 32 cols = N = 256

    // Cooperative coalesced load of the 32 x K A-tile
    for (int i = threadIdx.x; i < 32 * K; i += 256) {
        int r = i / K, c = i - r * K;
        As[r * stride + c] = A[(long long)(m0 + r) * K + c];
    }
    __syncthreads();

    // ISA fp32 fragment layout:
    //  A 16x4: lanes 0-15 -> M=lane, {K=0,K=1}; lanes 16-31 -> M=lane-16, {K=2,K=3}
    //  B 4x16: lanes 0-15 -> N=lane, {K=0,K=1}; lanes 16-31 -> N=lane-16, {K=2,K=3}
    const int lrow  = lane & 15;
    const int khalf = (lane >> 4) << 1;      // 0 or 2
    const float* arow0 = &As[lrow * stride];        // rows m0+0  .. m0+15
    const float* arow1 = &As[(16 + lrow) * stride]; // rows m0+16 .. m0+31

    v8f acc00 = {}, acc01 = {}, acc10 = {}, acc11 = {};
    for (int k0 = 0; k0 < K; k0 += 4) {
        v2f a0, a1, b0, b1;
        a0.x = arow0[k0 + khalf];
        a0.y = arow0[k0 + khalf + 1];
        a1.x = arow1[k0 + khalf];
        a1.y = arow1[k0 + khalf + 1];
        const float* wp = W + (long long)(k0 + khalf) * N + n0 + lrow;
        b0.x = wp[0];
        b0.y = wp[N];
        b1.x = wp[16];
        b1.y = wp[N + 16];
        acc00 = __builtin_amdgcn_wmma_f32_16x16x4_f32(false, a0, false, b0,
                                                      (short)0, acc00, false, false);
        acc01 = __builtin_amdgcn_wmma_f32_16x16x4_f32(false, a0, false, b1,
                                                      (short)0, acc01, false, false);
        acc10 = __builtin_amdgcn_wmma_f32_16x16x4_f32(false, a1, false, b0,
                                                      (short)0, acc10, false, false);
        acc11 = __builtin_amdgcn_wmma_f32_16x16x4_f32(false, a1, false, b1,
                                                      (short)0, acc11, false, false);
    }

    // C/D layout: VGPR r -> row r (lanes 0-15) or 8+r (lanes 16-31), col = lane%16
    const int col0   = n0 + lrow;
    const int col1   = col0 + 16;
    const int rbase0 = m0 + ((lane >> 4) << 3);
    const int rbase1 = rbase0 + 16;
    const float bv0 = bias ? bias[col0] : 0.0f;
    const float bv1 = bias ? bias[col1] : 0.0f;
#pragma unroll
    for (int r = 0; r < 8; ++r) {
        float v00 = acc00[r] + bv0, v01 = acc01[r] + bv1;
        float v10 = acc10[r] + bv0, v11 = acc11[r] + bv1;
        if (relu) {
            v00 = fmaxf(v00, 0.0f); v01 = fmaxf(v01, 0.0f);
            v10 = fmaxf(v10, 0.0f); v11 = fmaxf(v11, 0.0f);
        }
        C[(long long)(rbase0 + r) * N + col0] = v00;
        C[(long long)(rbase0 + r) * N + col1] = v01;
        C[(long long)(rbase1 + r) * N + col0] = v10;
        C[(long long)(rbase1 + r) * N + col1] = v11;
    }
}

// ---------------------------------------------------------------------------
// agg init: self-loop term + bias:  agg[n,f] = xw[n,f]*dinv[n]^2 + b[f]
// ---------------------------------------------------------------------------
__global__ __launch_bounds__(256) void agg_init_kernel(
    const float* __restrict__ xw, const float* __restrict__ dinv,
    const float* __restrict__ bias, float* __restrict__ agg, int Nn, int H) {
    int i = blockIdx.x * blockDim.x + threadIdx.x;
    if (i >= Nn * H) return;
    int n = i / H, f = i - n * H;
    float d = dinv[n];
    agg[i] = xw[i] * d * d + bias[f];
}

// ---------------------------------------------------------------------------
// Edge scatter: agg[dst] += xw[src] * dinv[src]*dinv[dst]
// Thread = (edge, 4-feature chunk). Consecutive lanes -> consecutive features
// of the same edge -> coalesced float4 gather, clustered atomics.
// ---------------------------------------------------------------------------
__global__ __launch_bounds__(256) void edge_scatter_kernel(
    const float* __restrict__ xw, const int* __restrict__ src,
    const int* __restrict__ dst, const float* __restrict__ dinv,
    float* __restrict__ agg, int E, int Hc /* = H/4 */) {
    int i = blockIdx.x * blockDim.x + threadIdx.x;
    if (i >= E * Hc) return;
    int e = i / Hc;
    int c = (i - e * Hc) << 2;
    int s = src[e], d = dst[e];
    float coeff = dinv[s] * dinv[d];
    const float4 v = *reinterpret_cast<const float4*>(xw + (long long)s * (Hc << 2) + c);
    float* out = agg + (long long)d * (Hc << 2) + c;
    atomicAdd(out + 0, v.x * coeff);
    atomicAdd(out + 1, v.y * coeff);
    atomicAdd(out + 2, v.z * coeff);
    atomicAdd(out + 3, v.w * coeff);
}

// ---------------------------------------------------------------------------
// BatchNorm stats: per-feature sum & sumsq. blockDim.x == H (256).
// Thread owns feature f, strides over rows -> fully coalesced row reads.
// ---------------------------------------------------------------------------
__global__ __launch_bounds__(256) void bn_stats_kernel(
    const float* __restrict__ h, float* __restrict__ stats, int Nn, int H) {
    int f = threadIdx.x;
    float s = 0.0f, sq = 0.0f;
    for (int n = blockIdx.x; n < Nn; n += gridDim.x) {
        float v = h[(long long)n * H + f];
        s += v;
        sq += v * v;
    }
    atomicAdd(&stats[f], s);
    atomicAdd(&stats[H + f], sq);
}

__global__ __launch_bounds__(256) void bn_norm_relu_kernel(
    float* __restrict__ h, const float* __restrict__ stats,
    const float* __restrict__ g, const float* __restrict__ be,
    int Nn, int H, float eps) {
    int i = blockIdx.x * blockDim.x + threadIdx.x;
    if (i >= Nn * H) return;
    int f = i & (H - 1);
    float invN = 1.0f / (float)Nn;
    float m   = stats[f] * invN;
    float var = stats[H + f] * invN - m * m;
    float sc  = rsqrtf(var + eps) * g[f];
    h[i] = fmaxf((h[i] - m) * sc + be[f], 0.0f);
}

// ---------------------------------------------------------------------------
// global_add_pool: pooled[batch[n], f] += h[n, f]
// ---------------------------------------------------------------------------
__global__ __launch_bounds__(256) void pool_kernel(
    const float* __restrict__ h, const int* __restrict__ batch,
    float* __restrict__ pooled, int Nn, int H) {
    int i = blockIdx.x * blockDim.x + threadIdx.x;
    if (i >= Nn * H) return;
    int n = i / H, f = i - n * H;
    atomicAdd(&pooled[(long long)batch[n] * H + f], h[i]);
}

// ---------------------------------------------------------------------------
// Final GEMV: out[g] = dot(t[g,:], w) + b0. One wave32 per row, shuffle reduce.
// ---------------------------------------------------------------------------
__global__ __launch_bounds__(256) void gemv_final_kernel(
    const float* __restrict__ t, const float* __restrict__ w,
    const float* __restrict__ b, float* __restrict__ out, int Gn, int H) {
    int g    = (blockIdx.x * blockDim.x + threadIdx.x) >> 5;
    int lane = threadIdx.x & 31;
    if (g >= Gn) return;
    float s = 0.0f;
    for (int f = lane; f < H; f += 32) s += t[(long long)g * H + f] * w[f];
#pragma unroll
    for (int off = 16; off > 0; off >>= 1) s += __shfl_down(s, off, 32);
    if (lane == 0) out[g] = s + b[0];
}

// ---------------------------------------------------------------------------
// Orchestration
// ---------------------------------------------------------------------------
static inline int cdiv(long long a, int b) { return (int)((a + b - 1) / b); }

extern "C" void kernel_launch(void* const* d_in, const int* in_sizes, int n_in,
                              void* d_out, int out_size, void* d_ws, size_t ws_size,
                              hipStream_t stream) {
    const float* x     = (const float*)d_in[0];
    const int*   ei    = (const int*)d_in[1];
    const int*   batch = (const int*)d_in[2];
    const float* W1 = (const float*)d_in[3];  const float* b1 = (const float*)d_in[4];
    const float* W2 = (const float*)d_in[5];  const float* b2 = (const float*)d_in[6];
    const float* W3 = (const float*)d_in[7];  const float* b3 = (const float*)d_in[8];
    const float* g1 = (const float*)d_in[9];  const float* be1 = (const float*)d_in[10];
    const float* g2 = (const float*)d_in[11]; const float* be2 = (const float*)d_in[12];
    const float* g3 = (const float*)d_in[13]; const float* be3 = (const float*)d_in[14];
    const float* LW1 = (const float*)d_in[15]; const float* Lb1 = (const float*)d_in[16];
    const float* LW2 = (const float*)d_in[17]; const float* Lb2 = (const float*)d_in[18];
    float* out = (float*)d_out;

    const int H    = in_sizes[4];            // 256
    const int F_IN = in_sizes[3] / H;        // 128
    const int Nn   = in_sizes[0] / F_IN;     // 100000
    const int E    = in_sizes[1] / 2;        // 300000
    const int Gn   = out_size;               // 2048
    const float eps = 1e-5f;

    const int* src = ei;
    const int* dst = ei + E;

    // Workspace carve-up
    float* ws     = (float*)d_ws;
    float* dinv   = ws;                      // Nn
    float* bufA   = dinv + ((Nn + 255) & ~255);          // Nn*H (xw)
    float* bufB   = bufA + (long long)Nn * H;            // Nn*H (agg / h)
    float* stats  = bufB + (long long)Nn * H;            // 2*H
    float* pooled = stats + 2 * H;                       // Gn*H
    float* thid   = pooled + (long long)Gn * H;          // Gn*H

    const int NT = 256;
    const long long NH = (long long)Nn * H;

    // degree -> dinv
    zero_kernel<<<cdiv(Nn, NT), NT, 0, stream>>>(dinv, Nn);
    deg_kernel<<<cdiv(E, NT), NT, 0, stream>>>(dst, dinv, E);
    dinv_kernel<<<cdiv(Nn, NT), NT, 0, stream>>>(dinv, Nn);

    const float* layer_in = x;
    int Kin = F_IN;
    const float* Wl[3]  = {W1, W2, W3};
    const float* bl[3]  = {b1, b2, b3};
    const float* gl[3]  = {g1, g2, g3};
    const float* bel[3] = {be1, be2, be3};

    for (int L = 0; L < 3; ++L) {
        // xw = in @ W  (WMMA fp32, 32-row blocks over full N)
        gemm_wmma_f32<<<dim3(Nn / 32), NT, 0, stream>>>(layer_in, Wl[L], nullptr, bufA,
                                                        Nn, H, Kin, 0);
        // agg = xw*dinv^2 + bias  (self loop), then scatter edges
        agg_init_kernel<<<cdiv(NH, NT), NT, 0, stream>>>(bufA, dinv, bl[L], bufB, Nn, H);
        edge_scatter_kernel<<<cdiv((long long)E * (H / 4), NT), NT, 0, stream>>>(
            bufA, src, dst, dinv, bufB, E, H / 4);
        // BatchNorm (training stats) + ReLU, in place
        zero_kernel<<<cdiv(2 * H, NT), NT, 0, stream>>>(stats, 2 * H);
        bn_stats_kernel<<<1024, H, 0, stream>>>(bufB, stats, Nn, H);
        bn_norm_relu_kernel<<<cdiv(NH, NT), NT, 0, stream>>>(bufB, stats, gl[L], bel[L],
                                                             Nn, H, eps);
        layer_in = bufB;
        Kin = H;
    }

    // global_add_pool
    zero_kernel<<<cdiv((long long)Gn * H, NT), NT, 0, stream>>>(pooled, Gn * H);
    pool_kernel<<<cdiv(NH, NT), NT, 0, stream>>>(bufB, batch, pooled, Nn, H);

    // MLP: relu(pooled @ LW1 + Lb1)  (WMMA fp32), then GEMV with LW2
    gemm_wmma_f32<<<dim3(Gn / 32), NT, 0, stream>>>(pooled, LW1, Lb1, thid, Gn, H, H, 1);
    gemv_final_kernel<<<cdiv((long long)Gn * 32, NT), NT, 0, stream>>>(thid, LW2, Lb2,
                                                                       out, Gn, H);
}